// CodaPrompt_3075196584395
// MI455X (gfx1250) — compile-verified
//
#include <hip/hip_runtime.h>

typedef float v2f __attribute__((ext_vector_type(2)));
typedef float v8f __attribute__((ext_vector_type(8)));

#define B_DIM 2048
#define D_DIM 768
#define E_DIM 5
#define K_DIM 100
#define L_DIM 8
#define EK    500          // E*K
#define EK_PAD 512         // padded leading dim for aq
#define N2    (L_DIM * D_DIM)   // 6144
#define HALF_N 3072        // (L/2)*D
#define EPSF  1e-12f

__device__ __forceinline__ v8f wmma4(v2f a, v2f b, v8f c) {
    // D = A(16x4) * B(4x16) + C, fp32 WMMA
    return __builtin_amdgcn_wmma_f32_16x16x4_f32(
        /*neg_a=*/false, a, /*neg_b=*/false, b,
        /*c_mod=*/(short)0, c, /*reuse_a=*/false, /*reuse_b=*/false);
}

// ---------------------------------------------------------------------------
// Kernel 1: per (e,k) row, compute W = As*Ks/max(||Ks||,eps) and A2 = As*As.
// One wave32 per row; shuffle reduction for the norm.
// ---------------------------------------------------------------------------
__global__ void __launch_bounds__(256) prep_rows(
        const float* __restrict__ Ks, const float* __restrict__ As,
        float* __restrict__ W, float* __restrict__ A2) {
    int gwid = (blockIdx.x * blockDim.x + threadIdx.x) >> 5;
    int lane = threadIdx.x & 31;
    if (gwid >= EK) return;
    const float* krow = Ks + (size_t)gwid * D_DIM;
    const float* arow = As + (size_t)gwid * D_DIM;
    float s = 0.f;
    for (int d = lane; d < D_DIM; d += 32) { float v = krow[d]; s += v * v; }
    #pragma unroll
    for (int m = 16; m >= 1; m >>= 1) s += __shfl_xor(s, m, 32);
    float inv = 1.0f / fmaxf(sqrtf(s), EPSF);
    for (int d = lane; d < D_DIM; d += 32) {
        float a = arow[d];
        W [(size_t)gwid * D_DIM + d] = a * krow[d] * inv;
        A2[(size_t)gwid * D_DIM + d] = a * a;
    }
}

// ---------------------------------------------------------------------------
// Kernel 2: x2 = x*x elementwise (float4 / b128 path)
// ---------------------------------------------------------------------------
__global__ void __launch_bounds__(256) square_x(
        const float4* __restrict__ x, float4* __restrict__ x2) {
    int i = blockIdx.x * blockDim.x + threadIdx.x;
    float4 v = x[i];
    v.x *= v.x; v.y *= v.y; v.z *= v.z; v.w *= v.w;
    x2[i] = v;
}

// ---------------------------------------------------------------------------
// Kernel 3: aq[b][ek] = (x @ W^T) / max(sqrt(x2 @ A2^T), eps)
// Wave computes 64(M) x 16(N); B fragments reused across 4 M-tiles.
// Padded N columns (ek in [500,512)) use a CLAMPED row pointer instead of a
// masked load: their accumulators hold garbage but are never stored, and in
// WMMA a B lane only feeds its own output column, so valid columns are exact.
// grid = (4, 32), block = 256 (8 waves).
// ---------------------------------------------------------------------------
__global__ void __launch_bounds__(256) gemm1(
        const float* __restrict__ x,  const float* __restrict__ x2,
        const float* __restrict__ W,  const float* __restrict__ A2,
        float* __restrict__ aq) {
    const int lane = threadIdx.x & 31;
    const int wid  = threadIdx.x >> 5;
    const int l15  = lane & 15;
    const int half = lane >> 4;
    const int m0   = (blockIdx.x * 8 + wid) * 64;     // 0..31 chunks of 64 rows
    const int ek   = blockIdx.y * 16 + l15;           // this lane's N column
    const bool ekv = (ek < EK);
    const int eks  = ekv ? ek : (EK - 1);             // clamped row for loads
    const float* wrow  = W  + (size_t)eks * D_DIM;
    const float* a2row = A2 + (size_t)eks * D_DIM;

    v8f cN0 = {}, cN1 = {}, cN2 = {}, cN3 = {};
    v8f cS0 = {}, cS1 = {}, cS2 = {}, cS3 = {};

    for (int kk = 0; kk < D_DIM; kk += 4) {
        const int col = kk + half * 2;                // K=col,col+1 per ISA A-layout
        v2f bW  = *(const v2f*)(wrow  + col);
        v2f bA  = *(const v2f*)(a2row + col);
        v2f ax0 = *(const v2f*)(x  + (size_t)(m0 +  0 + l15) * D_DIM + col);
        v2f ax1 = *(const v2f*)(x  + (size_t)(m0 + 16 + l15) * D_DIM + col);
        v2f ax2 = *(const v2f*)(x  + (size_t)(m0 + 32 + l15) * D_DIM + col);
        v2f ax3 = *(const v2f*)(x  + (size_t)(m0 + 48 + l15) * D_DIM + col);
        v2f as0 = *(const v2f*)(x2 + (size_t)(m0 +  0 + l15) * D_DIM + col);
        v2f as1 = *(const v2f*)(x2 + (size_t)(m0 + 16 + l15) * D_DIM + col);
        v2f as2 = *(const v2f*)(x2 + (size_t)(m0 + 32 + l15) * D_DIM + col);
        v2f as3 = *(const v2f*)(x2 + (size_t)(m0 + 48 + l15) * D_DIM + col);
        cN0 = wmma4(ax0, bW, cN0);  cS0 = wmma4(as0, bA, cS0);
        cN1 = wmma4(ax1, bW, cN1);  cS1 = wmma4(as1, bA, cS1);
        cN2 = wmma4(ax2, bW, cN2);  cS2 = wmma4(as2, bA, cS2);
        cN3 = wmma4(ax3, bW, cN3);  cS3 = wmma4(as3, bA, cS3);
    }

    if (ekv) {
        #define EPILOG1(CN, CS, I)                                             \
            _Pragma("unroll")                                                  \
            for (int r = 0; r < 8; ++r) {                                      \
                int m = m0 + (I) * 16 + r + half * 8;                          \
                aq[(size_t)m * EK_PAD + ek] =                                  \
                    CN[r] / fmaxf(sqrtf(CS[r]), EPSF);                         \
            }
        EPILOG1(cN0, cS0, 0)
        EPILOG1(cN1, cS1, 1)
        EPILOG1(cN2, cS2, 2)
        EPILOG1(cN3, cS3, 3)
        #undef EPILOG1
    }
}

// ---------------------------------------------------------------------------
// Kernel 4: out = aq[e] @ Ps[e]  (M=2048, N=6144, K=100), with the l-split
// folded into the store address. Block stages the 16x100 aq tile in LDS.
// grid = (128 Mtiles, 12 N-chunks of 512, 5 e), block = 256 (8 waves x 64N).
// ---------------------------------------------------------------------------
__global__ void __launch_bounds__(256) gemm2(
        const float* __restrict__ aq, const float* __restrict__ Ps,
        float* __restrict__ out) {
    __shared__ float lds_aq[16 * K_DIM];              // 6.4 KB, stride 100: bank-conflict free
    const int tid  = threadIdx.x;
    const int lane = tid & 31;
    const int wid  = tid >> 5;
    const int l15  = lane & 15;
    const int half = lane >> 4;
    const int m0   = blockIdx.x * 16;
    const int e    = blockIdx.z;

    for (int idx = tid; idx < 16 * K_DIM; idx += 256) {
        int r = idx / K_DIM;
        int k = idx - r * K_DIM;
        lds_aq[idx] = aq[(size_t)(m0 + r) * EK_PAD + e * K_DIM + k];
    }
    __syncthreads();

    const int nbase = blockIdx.y * 512 + wid * 64;
    const float* pe = Ps + (size_t)e * K_DIM * N2;
    v8f c0 = {}, c1 = {}, c2 = {}, c3 = {};

    for (int kk = 0; kk < K_DIM; kk += 4) {
        const int col = kk + half * 2;
        v2f a = *(const v2f*)(&lds_aq[l15 * K_DIM + col]);   // ds_load_b64
        const float* p0 = pe + (size_t)col * N2 + nbase + l15;
        // prefetch next K-step's B rows into cache (global_prefetch)
        __builtin_prefetch(p0 + (size_t)4 * N2, 0, 0);
        __builtin_prefetch(p0 + (size_t)5 * N2, 0, 0);
        v2f b0, b1, b2, b3;
        b0.x = p0[ 0];  b0.y = p0[N2 +  0];
        b1.x = p0[16];  b1.y = p0[N2 + 16];
        b2.x = p0[32];  b2.y = p0[N2 + 32];
        b3.x = p0[48];  b3.y = p0[N2 + 48];
        c0 = wmma4(a, b0, c0);
        c1 = wmma4(a, b1, c1);
        c2 = wmma4(a, b2, c2);
        c3 = wmma4(a, b3, c3);
    }

    // out shape (2, E, B, L/2, D); n = l*768+d ; keys: n<3072, values: n>=3072
    const size_t halfStride = (size_t)E_DIM * B_DIM * HALF_N;   // 31,457,280
    #pragma unroll
    for (int f = 0; f < 4; ++f) {
        int n = nbase + f * 16 + l15;
        int hsel = (n >= HALF_N) ? 1 : 0;
        size_t basep = (size_t)hsel * halfStride
                     + (size_t)e * B_DIM * HALF_N
                     + (size_t)(n - hsel * HALF_N);
        v8f c = (f == 0) ? c0 : (f == 1) ? c1 : (f == 2) ? c2 : c3;
        #pragma unroll
        for (int r = 0; r < 8; ++r) {
            int m = m0 + r + half * 8;
            out[basep + (size_t)m * HALF_N] = c[r];
        }
    }
}

// ---------------------------------------------------------------------------
extern "C" void kernel_launch(void* const* d_in, const int* in_sizes, int n_in,
                              void* d_out, int out_size, void* d_ws, size_t ws_size,
                              hipStream_t stream) {
    const float* x  = (const float*)d_in[0];   // (2048, 768)
    const float* Ks = (const float*)d_in[1];   // (5, 100, 768)
    const float* As = (const float*)d_in[2];   // (5, 100, 768)
    const float* Ps = (const float*)d_in[3];   // (5, 100, 8, 768)
    float* out = (float*)d_out;                // (2, 5, 2048, 4, 768)

    float* ws = (float*)d_ws;
    float* W   = ws;                                  // 500*768
    float* A2  = W  + (size_t)EK * D_DIM;             // 500*768
    float* x2  = A2 + (size_t)EK * D_DIM;             // 2048*768
    float* aqb = x2 + (size_t)B_DIM * D_DIM;          // 2048*512
    (void)in_sizes; (void)n_in; (void)out_size; (void)ws_size;

    // 500 waves -> 63 blocks of 8 waves
    prep_rows<<<dim3((EK * 32 + 255) / 256), 256, 0, stream>>>(Ks, As, W, A2);
    square_x <<<dim3((B_DIM * D_DIM / 4) / 256), 256, 0, stream>>>(
        (const float4*)x, (float4*)x2);
    gemm1    <<<dim3(4, 32), 256, 0, stream>>>(x, x2, W, A2, aqb);
    gemm2    <<<dim3(128, 12, 5), 256, 0, stream>>>(aqb, Ps, out);
}